// GVAE_71734543778151
// MI455X (gfx1250) — compile-verified
//
#include <hip/hip_runtime.h>
#include <math.h>

// ---------------------------------------------------------------------------
// GVAE forward for MI455X (gfx1250, wave32).
// All GEMMs use V_WMMA_F32_16X16X4_F32 (fp32 matrix path, exact vs reference).
// The N^2 decoder GEMM is algebraically factored: pairs@W1 = A[j]+B[i].
// ---------------------------------------------------------------------------

typedef float v2f __attribute__((ext_vector_type(2)));
typedef float v8f __attribute__((ext_vector_type(8)));

#define NN    768
#define LRELU(x) fmaxf((x), 0.01f * (x))

// ---------------------------------------------------------------------------
// Generic fp32 WMMA GEMM: C[M,N] = A[M,K] @ B[K,N] (+bias) (+sigmoid)
// One 16x16 C tile per wave, 8 waves per block. M%16==0, N%16==0, K%4==0.
// Fragment layouts per CDNA5 ISA 7.12.2:
//   A 16x4 : lane m=lane&15 ; VGPR0/1 hold K = k+2*(lane>=16) +{0,1}
//   B  4x16: lane n=lane&15 ; VGPR0/1 hold K = k+2*(lane>=16) +{0,1}
//   C 16x16: VGPR v -> row v + 8*(lane>=16), col = lane&15
// ---------------------------------------------------------------------------
__global__ __launch_bounds__(256)
void gemm_f32_wmma(const float* __restrict__ A, const float* __restrict__ B,
                   float* __restrict__ C, int M, int K, int N,
                   const float* __restrict__ bias, int act)
{
    const int lane = threadIdx.x & 31;
    const int wave = threadIdx.x >> 5;
    const int tn_count = N >> 4;
    const int tiles    = (M >> 4) * tn_count;
    const int tile     = blockIdx.x * 8 + wave;
    if (tile >= tiles) return;

    const int tm   = tile / tn_count;
    const int tn   = tile - tm * tn_count;
    const int half = lane >> 4;           // 0: lanes 0-15, 1: lanes 16-31
    const int l15  = lane & 15;
    const int mrow = tm * 16 + l15;
    const int ncol = tn * 16 + l15;

    const float* ap = A + (size_t)mrow * K + 2 * half;     // K-pair, 8B aligned
    const float* bp = B + (size_t)(2 * half) * N + ncol;

    v8f acc = {};
    for (int k = 0; k < K; k += 4) {
        v2f a = *(const v2f*)(ap + k);                     // A[m, k+2h .. +1]
        v2f b;
        b.x = bp[(size_t)k * N];                           // B[k+2h,   n]
        b.y = bp[(size_t)k * N + N];                       // B[k+2h+1, n]
        acc = __builtin_amdgcn_wmma_f32_16x16x4_f32(
                  false, a, false, b, (short)0, acc, false, false);
    }

    const float bv = bias ? bias[ncol] : 0.0f;
#pragma unroll
    for (int v = 0; v < 8; ++v) {
        const int row = tm * 16 + v + 8 * half;
        float val = acc[v] + bv;
        if (act == 1) val = 1.0f / (1.0f + __expf(-val));  // sigmoid epilogue
        C[(size_t)row * N + ncol] = val;
    }
}

// ---------------------------------------------------------------------------
// Degree / normalization
// ---------------------------------------------------------------------------
__global__ void deg_init(float* __restrict__ deg, int n) {
    int i = blockIdx.x * blockDim.x + threadIdx.x;
    if (i < n) deg[i] = 1.0f;                               // self loop
}
__global__ void deg_edges(const int* __restrict__ dst, float* __restrict__ deg, int E) {
    int e = blockIdx.x * blockDim.x + threadIdx.x;
    if (e < E) atomicAdd(&deg[dst[e]], 1.0f);
}
__global__ void deg_rsqrt(const float* __restrict__ deg, float* __restrict__ dinv, int n) {
    int i = blockIdx.x * blockDim.x + threadIdx.x;
    if (i < n) dinv[i] = rsqrtf(deg[i]);
}

// ---------------------------------------------------------------------------
// GCN scatter: out = selfloop + bias, then atomic edge accumulation
// ---------------------------------------------------------------------------
__global__ void self_loop_bias(const float* __restrict__ tmp, const float* __restrict__ dinv,
                               const float* __restrict__ bias, float* __restrict__ out,
                               int Nn, int F)
{
    int idx = blockIdx.x * blockDim.x + threadIdx.x;
    if (idx >= Nn * F) return;
    int n = idx / F, f = idx - n * F;
    float di = dinv[n];
    out[idx] = tmp[idx] * di * di + bias[f];
}
__global__ void scatter_edges(const float* __restrict__ tmp, const int* __restrict__ src,
                              const int* __restrict__ dst, const float* __restrict__ dinv,
                              float* __restrict__ out, int F)
{
    const int e = blockIdx.x;                 // one block per edge (uniform s,d -> SMEM)
    const int s = src[e];
    const int d = dst[e];
    const float nrm = dinv[s] * dinv[d];
    const int f = threadIdx.x;
    atomicAdd(&out[(size_t)d * F + f], tmp[(size_t)s * F + f] * nrm);
}

// ---------------------------------------------------------------------------
// BatchNorm (population var, eps=1e-5) + optional leaky_relu, in place.
// One block per column, 256 threads over 768 rows.
// ---------------------------------------------------------------------------
__global__ __launch_bounds__(256)
void bn_lrelu(float* __restrict__ h, const float* __restrict__ g,
              const float* __restrict__ b, int M, int F, int do_lrelu)
{
    __shared__ float ssum[256];
    __shared__ float ssq[256];
    const int f = blockIdx.x;
    const int tid = threadIdx.x;
    float s = 0.0f, q = 0.0f;
    for (int r = tid; r < M; r += 256) {
        float v = h[(size_t)r * F + f];
        s += v; q += v * v;
    }
    ssum[tid] = s; ssq[tid] = q;
    __syncthreads();
    for (int off = 128; off > 0; off >>= 1) {
        if (tid < off) { ssum[tid] += ssum[tid + off]; ssq[tid] += ssq[tid + off]; }
        __syncthreads();
    }
    const float mean = ssum[0] / (float)M;
    const float var  = ssq[0] / (float)M - mean * mean;
    const float inv  = rsqrtf(var + 1e-5f) * g[f];
    const float sh   = b[f];
    for (int r = tid; r < M; r += 256) {
        float v = (h[(size_t)r * F + f] - mean) * inv + sh;
        if (do_lrelu) v = LRELU(v);
        h[(size_t)r * F + f] = v;
    }
}

// ---------------------------------------------------------------------------
// Reparameterization: mu/logvar out, z = mu + eps*exp(0.5*logvar)
// ---------------------------------------------------------------------------
__global__ void reparam(const float* __restrict__ h4, const float* __restrict__ eps,
                        float* __restrict__ mu_o, float* __restrict__ lv_o,
                        float* __restrict__ z, int Nn, int lat)
{
    int idx = blockIdx.x * blockDim.x + threadIdx.x;
    if (idx >= Nn * lat) return;
    int n = idx / lat, f = idx - n * lat;
    float mu = h4[(size_t)n * 2 * lat + f];
    float lv = h4[(size_t)n * 2 * lat + lat + f];
    mu_o[idx] = mu;
    lv_o[idx] = lv;
    z[idx] = mu + eps[idx] * __expf(0.5f * lv);
}

// ---------------------------------------------------------------------------
// Pairwise decoder: adj[i,j] = sigmoid( sum_h lrelu(A[j,h]+B[i,h]) * w2[h] + b2 )
// (dec b1 is pre-folded into A). 32x32 output tile per block; A/B tiles in LDS
// with stride 257 (conflict-free), w2 broadcast from LDS. 4 pairs per thread.
// ---------------------------------------------------------------------------
#define PT   32
#define PSTR 257
__global__ __launch_bounds__(256)
void pair_decoder(const float* __restrict__ Adec, const float* __restrict__ Bdec,
                  const float* __restrict__ w2, const float* __restrict__ b2,
                  float* __restrict__ adj, int Nn, int H)
{
    __shared__ float As[PT * PSTR];
    __shared__ float Bs[PT * PSTR];
    __shared__ float Ws[256];
    const int tid = threadIdx.x;
    const int nb  = Nn / PT;
    const int bi  = blockIdx.x / nb;
    const int bj  = blockIdx.x - bi * nb;
    const int i0  = bi * PT, j0 = bj * PT;

    // col = tid (fixed), row = k : coalesced global, clean LDS writes
#pragma unroll 4
    for (int k = 0; k < PT; ++k) {
        As[k * PSTR + tid] = Adec[(size_t)(j0 + k) * H + tid];
        Bs[k * PSTR + tid] = Bdec[(size_t)(i0 + k) * H + tid];
    }
    Ws[tid] = w2[tid];
    __syncthreads();

    const int ii = tid >> 3;              // 0..31  -> i row
    const int jb = (tid & 7) * 4;         // j base -> 4 consecutive j
    const float* bsp = &Bs[ii * PSTR];
    const float* a0 = &As[(jb + 0) * PSTR];
    const float* a1 = &As[(jb + 1) * PSTR];
    const float* a2 = &As[(jb + 2) * PSTR];
    const float* a3 = &As[(jb + 3) * PSTR];

    float acc0 = 0.f, acc1 = 0.f, acc2 = 0.f, acc3 = 0.f;
#pragma unroll 4
    for (int h = 0; h < H; ++h) {
        const float bv = bsp[h];          // broadcast within 8-lane groups
        const float w  = Ws[h];           // wave-uniform broadcast
        float t0 = a0[h] + bv; t0 = LRELU(t0);
        float t1 = a1[h] + bv; t1 = LRELU(t1);
        float t2 = a2[h] + bv; t2 = LRELU(t2);
        float t3 = a3[h] + bv; t3 = LRELU(t3);
        acc0 = fmaf(t0, w, acc0);
        acc1 = fmaf(t1, w, acc1);
        acc2 = fmaf(t2, w, acc2);
        acc3 = fmaf(t3, w, acc3);
    }
    const float bb = b2[0];
    float4 r;
    r.x = 1.0f / (1.0f + __expf(-(acc0 + bb)));
    r.y = 1.0f / (1.0f + __expf(-(acc1 + bb)));
    r.z = 1.0f / (1.0f + __expf(-(acc2 + bb)));
    r.w = 1.0f / (1.0f + __expf(-(acc3 + bb)));
    *(float4*)&adj[(size_t)(i0 + ii) * Nn + (j0 + jb)] = r;   // 16B aligned
}

// ---------------------------------------------------------------------------
// Host-side orchestration
// ---------------------------------------------------------------------------
extern "C" void kernel_launch(void* const* d_in, const int* in_sizes, int n_in,
                              void* d_out, int out_size, void* d_ws, size_t ws_size,
                              hipStream_t stream)
{
    const float *x, *eps, *decW1, *decb1, *decW2, *decb2;
    const float *gcnW[4], *gcnB[4], *bng[3], *bnb[3];
    const float *nfW1, *nfb1, *nfg1, *nfbe1, *nfW2, *nfb2, *nfg2, *nfbe2, *nfW3, *nfb3;
    const int* edge;
    auto F = [&](int i) { return (const float*)d_in[i]; };

    int Ecnt;
    if (in_sizes[0] == NN * 128) {
        // setup_inputs() insertion order: x, edge_index, params{gcn,bn,dec,nf}, eps
        x = F(0); edge = (const int*)d_in[1]; Ecnt = in_sizes[1] / 2;
        gcnW[0]=F(2);  gcnB[0]=F(3);  gcnW[1]=F(4);  gcnB[1]=F(5);
        gcnW[2]=F(6);  gcnB[2]=F(7);  gcnW[3]=F(8);  gcnB[3]=F(9);
        bng[0]=F(10);  bnb[0]=F(11);  bng[1]=F(12);  bnb[1]=F(13);
        bng[2]=F(14);  bnb[2]=F(15);
        decW1=F(16); decb1=F(17); decW2=F(18); decb2=F(19);
        nfW1=F(20); nfb1=F(21); nfg1=F(22); nfbe1=F(23);
        nfW2=F(24); nfb2=F(25); nfg2=F(26); nfbe2=F(27);
        nfW3=F(28); nfb3=F(29); eps=F(30);
    } else {
        // jax tree_leaves order (dict keys sorted at every level)
        edge = (const int*)d_in[0]; Ecnt = in_sizes[0] / 2; eps = F(1);
        bnb[0]=F(2); bng[0]=F(3); bnb[1]=F(4); bng[1]=F(5); bnb[2]=F(6); bng[2]=F(7);
        decW1=F(8); decW2=F(9); decb1=F(10); decb2=F(11);
        gcnW[0]=F(12); gcnB[0]=F(13); gcnW[1]=F(14); gcnB[1]=F(15);
        gcnW[2]=F(16); gcnB[2]=F(17); gcnW[3]=F(18); gcnB[3]=F(19);
        nfW1=F(20); nfW2=F(21); nfW3=F(22); nfb1=F(23); nfb2=F(24); nfb3=F(25);
        nfbe1=F(26); nfbe2=F(27); nfg1=F(28); nfg2=F(29);
        x = F(30);
    }
    const int* src = edge;
    const int* dst = edge + Ecnt;

    // workspace layout (floats)
    float* ws   = (float*)d_ws;
    float* deg  = ws;                       // 768   (pad to 1024)
    float* dinv = ws + 1024;                // 768   (pad to 1024)
    float* buf0 = ws + 2048;                // 768*256
    float* buf1 = buf0 + NN * 256;          // 768*256
    float* buf2 = buf1 + NN * 256;          // 768*256
    float* z    = buf2 + NN * 256;          // 768*64

    float* out   = (float*)d_out;
    float* adj   = out;                     // 768*768
    float* feats = out + NN * NN;           // 768*128
    float* mu_o  = feats + NN * 128;        // 768*64
    float* lv_o  = mu_o + NN * 64;          // 768*64

    auto gemm_grid = [](int M, int N) { return ((M >> 4) * (N >> 4) + 7) / 8; };

    // degree + normalization
    deg_init <<<(NN + 255) / 256, 256, 0, stream>>>(deg, NN);
    deg_edges<<<(Ecnt + 255) / 256, 256, 0, stream>>>(dst, deg, Ecnt);
    deg_rsqrt<<<(NN + 255) / 256, 256, 0, stream>>>(deg, dinv, NN);

    // GCN layers 0..2 (K: 128,256,256 -> 256), BN + leaky_relu
    const float* hin = x;
    int Kd = 128;
    for (int i = 0; i < 3; ++i) {
        gemm_f32_wmma<<<gemm_grid(NN, 256), 256, 0, stream>>>(hin, gcnW[i], buf0,
                                                              NN, Kd, 256, nullptr, 0);
        self_loop_bias<<<NN * 256 / 256, 256, 0, stream>>>(buf0, dinv, gcnB[i], buf1, NN, 256);
        scatter_edges<<<Ecnt, 256, 0, stream>>>(buf0, src, dst, dinv, buf1, 256);
        bn_lrelu<<<256, 256, 0, stream>>>(buf1, bng[i], bnb[i], NN, 256, 1);
        hin = buf1; Kd = 256;
    }
    // GCN layer 3 (256 -> 128), no BN
    gemm_f32_wmma<<<gemm_grid(NN, 128), 256, 0, stream>>>(buf1, gcnW[3], buf0,
                                                          NN, 256, 128, nullptr, 0);
    self_loop_bias<<<NN * 128 / 256, 256, 0, stream>>>(buf0, dinv, gcnB[3], buf2, NN, 128);
    scatter_edges<<<Ecnt, 128, 0, stream>>>(buf0, src, dst, dinv, buf2, 128);

    // reparameterize (writes mu/logvar outputs, z to ws)
    reparam<<<NN * 64 / 256, 256, 0, stream>>>(buf2, eps, mu_o, lv_o, z, NN, 64);

    // decoder factorization: A = z @ W1[:64] + b1 ; B = z @ W1[64:]
    gemm_f32_wmma<<<gemm_grid(NN, 256), 256, 0, stream>>>(z, decW1,            buf0,
                                                          NN, 64, 256, decb1, 0);
    gemm_f32_wmma<<<gemm_grid(NN, 256), 256, 0, stream>>>(z, decW1 + 64 * 256, buf1,
                                                          NN, 64, 256, nullptr, 0);
    pair_decoder<<<(NN / PT) * (NN / PT), 256, 0, stream>>>(buf0, buf1, decW2, decb2,
                                                            adj, NN, 256);

    // node-feature decoder
    gemm_f32_wmma<<<gemm_grid(NN, 256), 256, 0, stream>>>(z, nfW1, buf0, NN, 64, 256, nfb1, 0);
    bn_lrelu<<<256, 256, 0, stream>>>(buf0, nfg1, nfbe1, NN, 256, 1);
    gemm_f32_wmma<<<gemm_grid(NN, 256), 256, 0, stream>>>(buf0, nfW2, buf2, NN, 256, 256, nfb2, 0);
    bn_lrelu<<<256, 256, 0, stream>>>(buf2, nfg2, nfbe2, NN, 256, 1);
    gemm_f32_wmma<<<gemm_grid(NN, 128), 256, 0, stream>>>(buf2, nfW3, feats, NN, 256, 128, nfb3, 1);

    (void)n_in; (void)out_size; (void)ws_size;
}